// BertRNNCRFDecoder_51221779972190
// MI455X (gfx1250) — compile-verified
//
#include <hip/hip_runtime.h>
#include <hip/hip_bf16.h>
#include <math.h>

// Problem constants (from the reference)
#define BB 64
#define TT 512
#define DD 1024
#define KK 48

typedef float v2f __attribute__((ext_vector_type(2)));
typedef float v8f __attribute__((ext_vector_type(8)));
typedef int   v4i __attribute__((ext_vector_type(4)));

#define TILE_D 256
#define SW_STRIDE 258   // 258 % 64 == 2 -> conflict-free B-fragment reads
#define NCHUNK (DD / TILE_D)

#if defined(__gfx1250__) && __has_builtin(__builtin_amdgcn_global_load_async_to_lds_b128)
#define USE_ASYNC_LDS 1
typedef __attribute__((address_space(1))) v4i* gv4i_p;   // global b128 unit
typedef __attribute__((address_space(3))) v4i* lv4i_p;   // LDS b128 unit
#else
#define USE_ASYNC_LDS 0
#endif

// ---------------------------------------------------------------------------
// Kernel 1: logits[(b*T+t)*48 + k] = hiddens row . W[k,:] + b[k]
// Grid: 256 blocks x 256 threads (8 waves). Each wave: one 16(M) x 48(N) tile.
// K-dim (D=1024) consumed in 4-wide steps with V_WMMA_F32_16X16X4_F32.
// W chunks staged into LDS; on gfx1250 this is double-buffered via
// GLOBAL_LOAD_ASYNC_TO_LDS_B128 + s_wait_asynccnt so the copy of chunk c+1
// overlaps the WMMA work on chunk c.
// ---------------------------------------------------------------------------
__global__ __launch_bounds__(256) void gemm_logits_kernel(
    const float* __restrict__ H,      // [32768,1024]
    const float* __restrict__ W,      // [48,1024]
    const float* __restrict__ bias,   // [48]
    float* __restrict__ logits)       // [32768,48]
{
    const int tid  = threadIdx.x;
    const int wave = tid >> 5;
    const int lane = tid & 31;
    const int half = lane >> 4;        // 0: K pair {0,1}, 1: K pair {2,3}
    const int mn   = lane & 15;        // M index for A, N index for B/C/D
    const int kpair = half * 2;

    const int mbase = blockIdx.x * 128 + wave * 16;
    const float* hrow = H + (size_t)(mbase + mn) * DD;

    v8f acc0 = {}; v8f acc1 = {}; v8f acc2 = {};

#if USE_ASYNC_LDS
    // Double-buffered W staging: async copy of chunk c+1 overlaps WMMA on c.
    __shared__ float sW[2][KK * SW_STRIDE];

    // Issue async copy of one W chunk [48 x 256] into LDS buffer `buf`.
    auto stage_async = [&](int c0, int buf) {
        #pragma unroll
        for (int i = tid; i < KK * (TILE_D / 4); i += 256) {
            const int r  = i >> 6;           // / (TILE_D/4)
            const int c4 = (i & 63) * 4;
            __builtin_amdgcn_global_load_async_to_lds_b128(
                (gv4i_p)(W + (size_t)r * DD + c0 + c4),
                (lv4i_p)(&sW[buf][r * SW_STRIDE + c4]),
                0, 0);
        }
    };

    stage_async(0, 0);
    asm volatile("s_wait_asynccnt 0x0" ::: "memory");
    __syncthreads();

    for (int c = 0; c < NCHUNK; ++c) {
        const int c0 = c * TILE_D;
        if (c + 1 < NCHUNK)
            stage_async(c0 + TILE_D, (c + 1) & 1);   // overlap with compute

        if (c0 + TILE_D < DD)
            __builtin_prefetch(hrow + c0 + TILE_D, 0, 0);  // global_prefetch_b8

        const float* sWc = &sW[c & 1][0];
        #pragma unroll 4
        for (int kk = 0; kk < TILE_D; kk += 4) {
            v2f a  = *(const v2f*)(hrow + c0 + kk + kpair);
            v2f b0 = *(const v2f*)(&sWc[(mn +  0) * SW_STRIDE + kk + kpair]);
            v2f b1 = *(const v2f*)(&sWc[(mn + 16) * SW_STRIDE + kk + kpair]);
            v2f b2 = *(const v2f*)(&sWc[(mn + 32) * SW_STRIDE + kk + kpair]);
            acc0 = __builtin_amdgcn_wmma_f32_16x16x4_f32(
                false, a, false, b0, (short)0, acc0, false, false);
            acc1 = __builtin_amdgcn_wmma_f32_16x16x4_f32(
                false, a, false, b1, (short)0, acc1, false, false);
            acc2 = __builtin_amdgcn_wmma_f32_16x16x4_f32(
                false, a, false, b2, (short)0, acc2, false, false);
        }
        // Chunk boundary: my async loads for c+1 are done, and all waves are
        // finished reading buffer c (so it can be overwritten at c+2).
        asm volatile("s_wait_asynccnt 0x0" ::: "memory");
        __syncthreads();
    }
#else
    // Fallback: synchronous staging through VGPRs, single buffer.
    __shared__ float sW[KK * SW_STRIDE];

    for (int c0 = 0; c0 < DD; c0 += TILE_D) {
        for (int i = tid; i < KK * (TILE_D / 4); i += 256) {
            const int r  = i >> 6;
            const int c4 = (i & 63) * 4;
            const float4 v = *(const float4*)(W + (size_t)r * DD + c0 + c4);
            float* dst = &sW[r * SW_STRIDE + c4];
            dst[0] = v.x; dst[1] = v.y; dst[2] = v.z; dst[3] = v.w;
        }
        __syncthreads();

        if (c0 + TILE_D < DD)
            __builtin_prefetch(hrow + c0 + TILE_D, 0, 0);

        #pragma unroll 4
        for (int kk = 0; kk < TILE_D; kk += 4) {
            v2f a  = *(const v2f*)(hrow + c0 + kk + kpair);
            v2f b0 = *(const v2f*)(&sW[(mn +  0) * SW_STRIDE + kk + kpair]);
            v2f b1 = *(const v2f*)(&sW[(mn + 16) * SW_STRIDE + kk + kpair]);
            v2f b2 = *(const v2f*)(&sW[(mn + 32) * SW_STRIDE + kk + kpair]);
            acc0 = __builtin_amdgcn_wmma_f32_16x16x4_f32(
                false, a, false, b0, (short)0, acc0, false, false);
            acc1 = __builtin_amdgcn_wmma_f32_16x16x4_f32(
                false, a, false, b1, (short)0, acc1, false, false);
            acc2 = __builtin_amdgcn_wmma_f32_16x16x4_f32(
                false, a, false, b2, (short)0, acc2, false, false);
        }
        __syncthreads();
    }
#endif

    // Epilogue: C/D layout -> VGPR v holds row (v + 8*half), col = mn
    const float bn0 = bias[mn];
    const float bn1 = bias[mn + 16];
    const float bn2 = bias[mn + 32];
    #pragma unroll
    for (int v = 0; v < 8; ++v) {
        const int row = mbase + v + 8 * half;
        float* out = logits + (size_t)row * KK;
        out[mn]      = acc0[v] + bn0;
        out[mn + 16] = acc1[v] + bn1;
        out[mn + 32] = acc2[v] + bn2;
    }
}

// ---------------------------------------------------------------------------
// Kernel 2: CRF log-partition (forward logsumexp scan) + gold score.
// One block per batch element, 64 threads (lane j < 48 owns next-state j).
// ---------------------------------------------------------------------------
__global__ __launch_bounds__(64) void crf_forward_kernel(
    const float* __restrict__ logits,   // [B*T,48]
    const int*   __restrict__ mask,     // [B,T]
    const int*   __restrict__ labels,   // [B,T]
    const float* __restrict__ startT,   // [48]
    const float* __restrict__ endT,     // [48]
    const float* __restrict__ trans,    // [48,48]
    float* __restrict__ logZ,           // [B]
    float* __restrict__ score)          // [B]
{
    __shared__ float sTrans[KK * KK];
    __shared__ float sAlpha[KK];
    __shared__ float sNext[KK];

    const int b   = blockIdx.x;
    const int tid = threadIdx.x;
    const float* em = logits + (size_t)b * TT * KK;

    for (int i = tid; i < KK * KK; i += 64) sTrans[i] = trans[i];
    if (tid < KK) sAlpha[tid] = startT[tid] + em[tid];
    __syncthreads();

    for (int t = 1; t < TT; ++t) {
        const float* e = em + (size_t)t * KK;
        const int mi = mask[b * TT + t];
        if (tid < KK) {
            float mx = -1e30f;
            #pragma unroll 8
            for (int i = 0; i < KK; ++i)
                mx = fmaxf(mx, sAlpha[i] + sTrans[i * KK + tid]);
            float s = 0.0f;
            #pragma unroll 8
            for (int i = 0; i < KK; ++i)
                s += __expf(sAlpha[i] + sTrans[i * KK + tid] - mx);
            const float nxt = mx + __logf(s) + e[tid];
            sNext[tid] = (mi > 0) ? nxt : sAlpha[tid];
        }
        __syncthreads();
        if (tid < KK) sAlpha[tid] = sNext[tid];
        __syncthreads();
    }

    if (tid == 0) {
        // logZ = logsumexp_j(alpha[j] + end[j])
        float mx = -1e30f;
        for (int j = 0; j < KK; ++j) mx = fmaxf(mx, sAlpha[j] + endT[j]);
        float s = 0.0f;
        for (int j = 0; j < KK; ++j) s += __expf(sAlpha[j] + endT[j] - mx);
        logZ[b] = mx + __logf(s);

        // gold score (sequential, matches reference semantics incl. mask)
        int prev = labels[b * TT];
        float sc = startT[prev] + em[prev];
        for (int t = 1; t < TT; ++t) {
            const int tg = labels[b * TT + t];
            const int mi = mask[b * TT + t];
            const float st = sTrans[prev * KK + tg] + em[(size_t)t * KK + tg];
            sc += st * (float)mi;
            prev = (mi > 0) ? tg : prev;
        }
        score[b] = sc + endT[prev];
    }
}

// ---------------------------------------------------------------------------
// Kernel 3: Viterbi forward (max/argmax scan) + backtrack -> pred tags.
// One block per batch element, 64 threads.
// ---------------------------------------------------------------------------
__global__ __launch_bounds__(64) void crf_viterbi_kernel(
    const float* __restrict__ logits,
    const int*   __restrict__ mask,
    const float* __restrict__ startT,
    const float* __restrict__ endT,
    const float* __restrict__ trans,
    unsigned char* __restrict__ hist,   // [(T-1), B, 48]
    float* __restrict__ pred)           // [B,T] tags written as float
{
    __shared__ float sTrans[KK * KK];
    __shared__ float sScore[KK];
    __shared__ float sNext[KK];

    const int b   = blockIdx.x;
    const int tid = threadIdx.x;
    const float* em = logits + (size_t)b * TT * KK;

    for (int i = tid; i < KK * KK; i += 64) sTrans[i] = trans[i];
    if (tid < KK) sScore[tid] = startT[tid] + em[tid];
    __syncthreads();

    for (int t = 1; t < TT; ++t) {
        const float* e = em + (size_t)t * KK;
        const int mi = mask[b * TT + t];
        if (tid < KK) {
            float best = -1e30f;
            int bi = 0;
            #pragma unroll 8
            for (int i = 0; i < KK; ++i) {
                const float v = sScore[i] + sTrans[i * KK + tid];
                if (v > best) { best = v; bi = i; }
            }
            hist[((size_t)(t - 1) * BB + b) * KK + tid] = (unsigned char)bi;
            const float nxt = best + e[tid];
            sNext[tid] = (mi > 0) ? nxt : sScore[tid];
        }
        __syncthreads();
        if (tid < KK) sScore[tid] = sNext[tid];
        __syncthreads();
    }

    if (tid == 0) {
        float best = -1e30f;
        int tag = 0;
        for (int j = 0; j < KK; ++j) {
            const float v = sScore[j] + endT[j];
            if (v > best) { best = v; tag = j; }
        }
        pred[b * TT + (TT - 1)] = (float)tag;
        for (int t = TT - 2; t >= 0; --t) {
            const int mi = mask[b * TT + t + 1];
            const int prev = hist[((size_t)t * BB + b) * KK + tag];
            if (mi > 0) tag = prev;
            pred[b * TT + t] = (float)tag;
        }
    }
}

// ---------------------------------------------------------------------------
// Kernel 4: loss = -mean(score - logZ); written after the pred block.
// ---------------------------------------------------------------------------
__global__ __launch_bounds__(64) void crf_loss_kernel(
    const float* __restrict__ logZ,
    const float* __restrict__ score,
    float* __restrict__ out_loss)
{
    __shared__ float red[64];
    const int tid = threadIdx.x;
    red[tid] = (tid < BB) ? (score[tid] - logZ[tid]) : 0.0f;
    __syncthreads();
    for (int s = 32; s > 0; s >>= 1) {
        if (tid < s) red[tid] += red[tid + s];
        __syncthreads();
    }
    if (tid == 0) out_loss[0] = -red[0] / (float)BB;
}

// ---------------------------------------------------------------------------
extern "C" void kernel_launch(void* const* d_in, const int* in_sizes, int n_in,
                              void* d_out, int out_size, void* d_ws, size_t ws_size,
                              hipStream_t stream) {
    const float* hiddens = (const float*)d_in[0];   // [B,T,D]
    const int*   mask    = (const int*)  d_in[1];   // [B,T]
    const int*   labels  = (const int*)  d_in[2];   // [B,T]
    const float* W       = (const float*)d_in[3];   // [K,D]
    const float* bvec    = (const float*)d_in[4];   // [K]
    const float* startT  = (const float*)d_in[5];   // [K]
    const float* endT    = (const float*)d_in[6];   // [K]
    const float* trans   = (const float*)d_in[7];   // [K,K]

    float* out = (float*)d_out;          // [B*T] pred tags, then [1] loss
    char*  ws  = (char*)d_ws;

    // Workspace layout
    float* logits = (float*)ws;                               // 32768*48 f32
    size_t off = (size_t)BB * TT * KK * sizeof(float);
    float* logZ   = (float*)(ws + off); off += BB * sizeof(float);
    float* score  = (float*)(ws + off); off += BB * sizeof(float);
    off = (off + 255) & ~(size_t)255;
    unsigned char* hist = (unsigned char*)(ws + off);         // (T-1)*B*48 u8

    // 1) projection GEMM via V_WMMA_F32_16X16X4_F32 (+ async W staging)
    gemm_logits_kernel<<<dim3(256), dim3(256), 0, stream>>>(hiddens, W, bvec, logits);
    // 2) log-partition + gold score
    crf_forward_kernel<<<dim3(BB), dim3(64), 0, stream>>>(
        logits, mask, labels, startT, endT, trans, logZ, score);
    // 3) Viterbi decode -> pred tags (floats) into d_out[0 .. B*T)
    crf_viterbi_kernel<<<dim3(BB), dim3(64), 0, stream>>>(
        logits, mask, startT, endT, trans, hist, out);
    // 4) loss -> d_out[B*T]
    crf_loss_kernel<<<dim3(1), dim3(64), 0, stream>>>(logZ, score, out + (size_t)BB * TT);
}